// MaskedStableExpressivePE_44573170597934
// MI455X (gfx1250) — compile-verified
//
#include <hip/hip_runtime.h>

typedef __attribute__((ext_vector_type(2))) float v2f;
typedef __attribute__((ext_vector_type(8))) float v8f;

#define B_GRAPH 128
#define NNODE   256
#define D_PE    16
#define M_PSI   4
#define H_PSI   32
#define H_P     16
#define H_G     64
#define N_SUM   (B_GRAPH * NNODE)
#define E_EDGE  (N_SUM * 16)

__device__ __forceinline__ v8f wmma_f32_k4(v2f a, v2f b, v8f c) {
  return __builtin_amdgcn_wmma_f32_16x16x4_f32(false, a, false, b, (short)0, c,
                                               false, false);
}

// async copy of 64 bytes (one 16-float row) global -> LDS, no VGPR staging.
// INST_OFFSET is applied to both the LDS and global side (ISA 08 §4.4).
__device__ __forceinline__ void async_copy_row64(unsigned lds_byte_off,
                                                 const float* gsrc) {
  asm volatile(
      "global_load_async_to_lds_b128 %0, %1, off\n\t"
      "global_load_async_to_lds_b128 %0, %1, off offset:16\n\t"
      "global_load_async_to_lds_b128 %0, %1, off offset:32\n\t"
      "global_load_async_to_lds_b128 %0, %1, off offset:48"
      :: "v"(lds_byte_off), "v"(gsrc) : "memory");
}

__device__ __forceinline__ void async_copy_16B(unsigned lds_byte_off,
                                               const float* gsrc) {
  asm volatile("global_load_async_to_lds_b128 %0, %1, off"
               :: "v"(lds_byte_off), "v"(gsrc) : "memory");
}

__device__ __forceinline__ void wait_async0() {
  asm volatile("s_wait_asynccnt 0" ::: "memory");
}

// ---------------- zero workspace (agg + counts) ----------------
__global__ void k_zero(float* agg, int* cnt) {
  int i = blockIdx.x * blockDim.x + threadIdx.x;
  if (i < N_SUM * D_PE) agg[i] = 0.0f;
  if (i < B_GRAPH) cnt[i] = 0;
}

// ---------------- per-graph node counts ----------------
__global__ void k_count(const int* batch, int* cnt) {
  int i = blockIdx.x * blockDim.x + threadIdx.x;
  if (i < N_SUM) atomicAdd(&cnt[batch[i]], 1);
}

// ---------------- Z[b,d,m]: per-eigenvalue MLP, masked ----------------
__global__ void k_z(const float* __restrict__ Lambda,
                    const float* __restrict__ W1, const float* __restrict__ b1,
                    const float* __restrict__ W2, const float* __restrict__ b2,
                    const int* __restrict__ cnt, float* __restrict__ Z) {
  int b = blockIdx.x;
  int t = threadIdx.x;          // 64 threads: d = t/4, m = t%4
  int d = t >> 2, m = t & 3;
  float lam = Lambda[b * D_PE + d];
  float acc = 0.0f;
  #pragma unroll
  for (int h = 0; h < H_PSI; ++h) {
    float v = lam * W1[m * H_PSI + h] + b1[m * H_PSI + h];
    acc += fmaxf(v, 0.0f) * W2[m * H_PSI + h];
  }
  acc += b2[m];
  if (d >= cnt[b]) acc = 0.0f;
  Z[b * (D_PE * M_PSI) + t] = acc;
}

// ---------------- fused W = V diag(Z) V^T ; h1 = relu(W Wp1 + bp1) ; x = sum_k h1
__global__ void __launch_bounds__(256)
k_main(const float* __restrict__ V, const float* __restrict__ Z,
       const float* __restrict__ Wp1, const float* __restrict__ bp1,
       float* __restrict__ x) {
  __shared__ float Vl[NNODE * D_PE];          // 16 KB
  __shared__ float Zl[D_PE * M_PSI];          // 256 B, layout [d][m]
  __shared__ float Wl[8][16 * 16 * M_PSI];    // per-wave scratch, [n][k][m]

  int b = blockIdx.x;
  int t = threadIdx.x;
  int lane = t & 31, wave = t >> 5;

  // async fill of V_b: each thread copies one 16-float row straight to LDS
  {
    const float* gsrc = V + (size_t)b * NNODE * D_PE + (size_t)t * D_PE;
    unsigned ldsa = (unsigned)(size_t)(void*)&Vl[t * D_PE];
    async_copy_row64(ldsa, gsrc);
  }
  if (t < D_PE * M_PSI) Zl[t] = Z[b * (D_PE * M_PSI) + t];
  wait_async0();
  __syncthreads();

  int row   = lane & 15;
  int off   = (lane < 16) ? 0 : 2;
  int nbase = 8 * (lane >> 4);

  // B-operand for the Wp1 transform: B[m,h], h = lane%16, m = off/off+1
  v2f bw;
  bw.x = Wp1[off * H_P + row];
  bw.y = Wp1[(off + 1) * H_P + row];
  float bp1v = bp1[row];

  for (int nt = wave; nt < 16; nt += 8) {
    int n0 = nt * 16;

    // A-operand chunks: A[n,d] = V[n0+n, d] * Z[d, m]  (4 chunks of K=4, per m)
    v2f a[M_PSI][4];
    #pragma unroll
    for (int c = 0; c < 4; ++c) {
      int d0 = 4 * c + off;
      float va = Vl[(n0 + row) * D_PE + d0];
      float vb = Vl[(n0 + row) * D_PE + d0 + 1];
      #pragma unroll
      for (int m = 0; m < M_PSI; ++m) {
        a[m][c].x = va * Zl[d0 * M_PSI + m];
        a[m][c].y = vb * Zl[(d0 + 1) * M_PSI + m];
      }
    }

    v8f xacc;
    #pragma unroll
    for (int j = 0; j < 8; ++j) xacc[j] = 0.0f;

    for (int kt = 0; kt < 16; ++kt) {
      int k0 = kt * 16;
      // B-operand chunks: B[d,k] = V[k0+k, d]
      v2f bv[4];
      #pragma unroll
      for (int c = 0; c < 4; ++c) {
        int d0 = 4 * c + off;
        bv[c].x = Vl[(k0 + row) * D_PE + d0];
        bv[c].y = Vl[(k0 + row) * D_PE + d0 + 1];
      }
      // stage 1: Wm[n,k] = sum_d A B, 4 chained K=4 WMMAs per m
      v8f wm[M_PSI];
      #pragma unroll
      for (int m = 0; m < M_PSI; ++m) {
        v8f acc;
        #pragma unroll
        for (int j = 0; j < 8; ++j) acc[j] = 0.0f;
        #pragma unroll
        for (int c = 0; c < 4; ++c) acc = wmma_f32_k4(a[m][c], bv[c], acc);
        wm[m] = acc;
      }
      // spill accumulators to per-wave LDS in [n][k][m] order (m contiguous)
      #pragma unroll
      for (int m = 0; m < M_PSI; ++m) {
        #pragma unroll
        for (int j = 0; j < 8; ++j)
          Wl[wave][((nbase + j) * 16 + row) * M_PSI + m] = wm[m][j];
      }
      // stage 2: for each k in tile, P = relu(W[:,k,:] Wp1 + bp1); x += P
      #pragma unroll
      for (int k = 0; k < 16; ++k) {
        v2f a2;
        a2.x = Wl[wave][(row * 16 + k) * M_PSI + off];
        a2.y = Wl[wave][(row * 16 + k) * M_PSI + off + 1];
        v8f cini;
        #pragma unroll
        for (int j = 0; j < 8; ++j) cini[j] = bp1v;
        v8f p = wmma_f32_k4(a2, bw, cini);
        #pragma unroll
        for (int j = 0; j < 8; ++j) xacc[j] += fmaxf(p[j], 0.0f);
      }
    }
    // store x tile (C-layout: row = nbase+j, col = lane%16)
    #pragma unroll
    for (int j = 0; j < 8; ++j) {
      int n = n0 + nbase + j;
      x[((size_t)b * NNODE + n) * D_PE + row] = xacc[j];
    }
  }
}

// ---------------- GIN edge aggregation: agg[dst] += x[src] ----------------
__global__ void k_edge(const int* __restrict__ ei, const float* __restrict__ x,
                       float* __restrict__ agg) {
  int e = blockIdx.x * blockDim.x + threadIdx.x;
  if (e >= E_EDGE) return;
  int s = ei[e];
  int d = ei[E_EDGE + e];
  const float* xs = x + (size_t)s * D_PE;
  float* ad = agg + (size_t)d * D_PE;
  #pragma unroll
  for (int c = 0; c < D_PE; ++c) atomicAdd(&ad[c], xs[c]);
}

// ---------------- final MLP: relu(((1+eps)x + agg) Wg1 + bg1) Wg2 + bg2 ----
__global__ void __launch_bounds__(256)
k_gin(const float* __restrict__ x, const float* __restrict__ agg,
      const float* __restrict__ Wg1, const float* __restrict__ bg1,
      const float* __restrict__ Wg2, const float* __restrict__ bg2,
      const float* __restrict__ eps, float* __restrict__ out) {
  __shared__ float Wg1l[D_PE * H_G];   // 4 KB
  __shared__ float Wg2l[H_G * D_PE];   // 4 KB
  __shared__ float Xt[8][16 * 16];     // 8 KB
  __shared__ float H2[8][16 * H_G];    // 32 KB

  int t = threadIdx.x, lane = t & 31, wave = t >> 5;

  // async-stage weights: each array = 256 x 16B, one async b128 per thread
  async_copy_16B((unsigned)(size_t)(void*)&Wg1l[t * 4], Wg1 + t * 4);
  async_copy_16B((unsigned)(size_t)(void*)&Wg2l[t * 4], Wg2 + t * 4);

  float e1 = 1.0f + eps[0];
  int tile = blockIdx.x * 8 + wave;

  // load input tile: Xt[n][c] = (1+eps)*x + agg   (each lane: half a row)
  int rrow = lane >> 1, half = lane & 1;
  int node = tile * 16 + rrow;
  const float4* px = (const float4*)(x + (size_t)node * D_PE) + half * 2;
  const float4* pa = (const float4*)(agg + (size_t)node * D_PE) + half * 2;
  float4 xv0 = px[0], xv1 = px[1];
  float4 av0 = pa[0], av1 = pa[1];
  float* xd = &Xt[wave][rrow * 16 + half * 8];
  xd[0] = e1 * xv0.x + av0.x;  xd[1] = e1 * xv0.y + av0.y;
  xd[2] = e1 * xv0.z + av0.z;  xd[3] = e1 * xv0.w + av0.w;
  xd[4] = e1 * xv1.x + av1.x;  xd[5] = e1 * xv1.y + av1.y;
  xd[6] = e1 * xv1.z + av1.z;  xd[7] = e1 * xv1.w + av1.w;
  wait_async0();
  __syncthreads();

  int row = lane & 15;
  int off = (lane < 16) ? 0 : 2;
  int nbase = 8 * (lane >> 4);

  // GEMM1: H2 = relu(Xt(16x16) * Wg1(16x64) + bg1), 4 col-tiles x 4 K-chunks
  #pragma unroll
  for (int tc = 0; tc < 4; ++tc) {
    v8f acc;
    float bg = bg1[tc * 16 + row];
    #pragma unroll
    for (int j = 0; j < 8; ++j) acc[j] = bg;
    #pragma unroll
    for (int c = 0; c < 4; ++c) {
      int d0 = 4 * c + off;
      v2f aa; aa.x = Xt[wave][row * 16 + d0];
              aa.y = Xt[wave][row * 16 + d0 + 1];
      v2f bb; bb.x = Wg1l[d0 * H_G + tc * 16 + row];
              bb.y = Wg1l[(d0 + 1) * H_G + tc * 16 + row];
      acc = wmma_f32_k4(aa, bb, acc);
    }
    #pragma unroll
    for (int j = 0; j < 8; ++j)
      H2[wave][(nbase + j) * H_G + tc * 16 + row] = fmaxf(acc[j], 0.0f);
  }

  // GEMM2: out = H2(16x64) * Wg2(64x16) + bg2, 16 K-chunks
  v8f acc2;
  float bg2v = bg2[row];
  #pragma unroll
  for (int j = 0; j < 8; ++j) acc2[j] = bg2v;
  #pragma unroll
  for (int c = 0; c < 16; ++c) {
    int d0 = 4 * c + off;
    v2f aa; aa.x = H2[wave][row * H_G + d0];
            aa.y = H2[wave][row * H_G + d0 + 1];
    v2f bb; bb.x = Wg2l[d0 * D_PE + row];
            bb.y = Wg2l[(d0 + 1) * D_PE + row];
    acc2 = wmma_f32_k4(aa, bb, acc2);
  }
  #pragma unroll
  for (int j = 0; j < 8; ++j) {
    int n = tile * 16 + nbase + j;
    out[(size_t)n * D_PE + row] = acc2[j];
  }
}

extern "C" void kernel_launch(void* const* d_in, const int* in_sizes, int n_in,
                              void* d_out, int out_size, void* d_ws, size_t ws_size,
                              hipStream_t stream) {
  const float* Lambda = (const float*)d_in[0];
  const float* V      = (const float*)d_in[1];
  const float* psi_W1 = (const float*)d_in[2];
  const float* psi_b1 = (const float*)d_in[3];
  const float* psi_W2 = (const float*)d_in[4];
  const float* psi_b2 = (const float*)d_in[5];
  const float* Wp1    = (const float*)d_in[6];
  const float* bp1    = (const float*)d_in[7];
  const float* Wg1    = (const float*)d_in[8];
  const float* bg1    = (const float*)d_in[9];
  const float* Wg2    = (const float*)d_in[10];
  const float* bg2    = (const float*)d_in[11];
  const float* eps    = (const float*)d_in[12];
  const int* edge_index = (const int*)d_in[13];
  const int* batch      = (const int*)d_in[14];
  float* out = (float*)d_out;

  char* ws = (char*)d_ws;
  float* Zws  = (float*)ws;                                   // 32 KB
  int*   cnt  = (int*)(ws + 32 * 1024);                       // 512 B
  float* xws  = (float*)(ws + 64 * 1024);                     // 2 MB
  float* aggw = (float*)(ws + 64 * 1024 +
                         (size_t)N_SUM * D_PE * sizeof(float)); // 2 MB

  k_zero <<<(N_SUM * D_PE + 255) / 256, 256, 0, stream>>>(aggw, cnt);
  k_count<<<(N_SUM + 255) / 256, 256, 0, stream>>>(batch, cnt);
  k_z    <<<B_GRAPH, 64, 0, stream>>>(Lambda, psi_W1, psi_b1, psi_W2, psi_b2,
                                      cnt, Zws);
  k_main <<<B_GRAPH, 256, 0, stream>>>(V, Zws, Wp1, bp1, xws);
  k_edge <<<(E_EDGE + 255) / 256, 256, 0, stream>>>(edge_index, xws, aggw);
  k_gin  <<<N_SUM / (16 * 8), 256, 0, stream>>>(xws, aggw, Wg1, bg1, Wg2, bg2,
                                                eps, out);
}